// DeeperGCN_43843026157847
// MI455X (gfx1250) — compile-verified
//
#include <hip/hip_runtime.h>
#include <math.h>

typedef float v2f __attribute__((ext_vector_type(2)));
typedef float v8f __attribute__((ext_vector_type(8)));

#define HD __device__ __forceinline__

static constexpr int   Hc   = 64;      // hidden channels
static constexpr float GEPS = 1e-7f;   // GENConv message epsilon

// ---------------------------------------------------------------------------
// float atomic max via sign-split integer atomics (IEEE-754 order trick).
// ---------------------------------------------------------------------------
HD void atomicMaxF(float* addr, float val) {
    if (val >= 0.0f)
        atomicMax((int*)addr, __float_as_int(val));
    else
        atomicMin((unsigned int*)addr, __float_as_uint(val));
}

// ---------------------------------------------------------------------------
// WMMA fp32 GEMM: C[M,N] = A[M,K] @ B[K,N] + bias[N].
// One wave computes a 16x64 output slab = 4 WMMA tiles sharing one A fragment
// per k-step. Software-pipelined: loads for step k+4 are issued before the
// WMMAs of step k, so the s_wait before each WMMA covers loads with a full
// iteration of slack. Ragged N (classifier N=40): B addresses are clamped
// in-range; out-of-range columns produce garbage only in accumulator columns
// that are never stored, so no masking is needed and EXEC stays all-1s.
//
// V_WMMA_F32_16X16X4_F32 operand layout (ISA 7.12.2):
//   A 16x4 : lane l -> row m=l&15; VGPR0/1 hold K = 2*(l>>4) + {0,1}
//   B  4x16: lane l -> col n=l&15; VGPR0/1 hold K = 2*(l>>4) + {0,1}
//   C/D    : VGPR r, lane l -> (m = r + 8*(l>>4), n = l&15)
// ---------------------------------------------------------------------------
__global__ void gemm_bias_wmma(const float* __restrict__ A,
                               const float* __restrict__ B,
                               const float* __restrict__ bias,
                               float* __restrict__ C,
                               int M, int N, int K) {
    int wave    = (int)((blockIdx.x * blockDim.x + threadIdx.x) >> 5);
    int lane    = threadIdx.x & 31;
    int groupsN = (N + 63) >> 6;           // 64-column slabs
    int tm = wave / groupsN;
    int gn = wave - tm * groupsN;
    if (tm >= (M >> 4)) return;            // wave-uniform: EXEC stays full

    int mr   = lane & 15;
    int half = lane >> 4;                  // 0 or 1
    const float* Arow = A + (long)(tm * 16 + mr) * K;

    int ncol[4];
#pragma unroll
    for (int t = 0; t < 4; ++t) {
        int n   = gn * 64 + t * 16 + mr;
        ncol[t] = (n < N) ? n : (N - 1);   // clamped, always in-range
    }

    v8f acc[4] = {{0.f,0.f,0.f,0.f,0.f,0.f,0.f,0.f},
                  {0.f,0.f,0.f,0.f,0.f,0.f,0.f,0.f},
                  {0.f,0.f,0.f,0.f,0.f,0.f,0.f,0.f},
                  {0.f,0.f,0.f,0.f,0.f,0.f,0.f,0.f}};

    // ---- prologue: load k=0 fragments -------------------------------------
    int ka = half * 2;
    v2f a_cur;
    a_cur.x = Arow[ka];
    a_cur.y = Arow[ka + 1];
    v2f b_cur[4];
#pragma unroll
    for (int t = 0; t < 4; ++t) {
        b_cur[t].x = B[ka * N + ncol[t]];
        b_cur[t].y = B[(ka + 1) * N + ncol[t]];
    }

    // ---- pipelined main loop ----------------------------------------------
    for (int k = 0; k < K; k += 4) {
        int kn = (k + 4 < K) ? (k + 4) : k;      // last iter: harmless reload
        int kb = kn + half * 2;

        v2f a_nxt;
        a_nxt.x = Arow[kb];
        a_nxt.y = Arow[kb + 1];
        v2f b_nxt[4];
#pragma unroll
        for (int t = 0; t < 4; ++t) {
            b_nxt[t].x = B[kb * N + ncol[t]];
            b_nxt[t].y = B[(kb + 1) * N + ncol[t]];
        }

#pragma unroll
        for (int t = 0; t < 4; ++t)
            acc[t] = __builtin_amdgcn_wmma_f32_16x16x4_f32(
                         false, a_cur, false, b_cur[t], (short)0, acc[t], false, false);

        a_cur = a_nxt;
#pragma unroll
        for (int t = 0; t < 4; ++t) b_cur[t] = b_nxt[t];
    }

    // ---- epilogue: bias + guarded stores ----------------------------------
#pragma unroll
    for (int t = 0; t < 4; ++t) {
        int n = gn * 64 + t * 16 + mr;
        if (n < N) {
            float bv = bias[n];
#pragma unroll
            for (int r = 0; r < 8; ++r)
                C[(long)(tm * 16 + half * 8 + r) * N + n] = acc[t][r] + bv;
        }
    }
}

// ---------------------------------------------------------------------------
// Wave-per-row LayerNorm + ReLU.  D in {64, 128}; lane holds D/32 elements.
// ---------------------------------------------------------------------------
__global__ void ln_relu_kernel(const float* __restrict__ X,
                               const float* __restrict__ g,
                               const float* __restrict__ bvec,
                               float* __restrict__ Y,
                               int rows, int D) {
    int warp = threadIdx.x >> 5;
    int lane = threadIdx.x & 31;
    int row  = blockIdx.x * (blockDim.x >> 5) + warp;
    if (row >= rows) return;

    const float* x = X + (long)row * D;
    int   nper = D >> 5;                   // 2 or 4
    float vals[4];
    float s = 0.f, ss = 0.f;
    for (int j = 0; j < nper; ++j) {
        float v = x[lane + j * 32];
        vals[j] = v;
        s  += v;
        ss += v * v;
    }
    for (int off = 16; off; off >>= 1) {
        s  += __shfl_xor(s,  off);
        ss += __shfl_xor(ss, off);
    }
    float mean = s / (float)D;
    float var  = ss / (float)D - mean * mean;
    float rstd = rsqrtf(var + 1e-5f);

    float* y = Y + (long)row * D;
    for (int j = 0; j < nper; ++j) {
        int   idx = lane + j * 32;
        float v   = (vals[j] - mean) * rstd * g[idx] + bvec[idx];
        y[idx] = fmaxf(v, 0.f);
    }
}

// ---------------------------------------------------------------------------
// Edge phase (bandwidth/atomic bound: ~245 MB gather + E*H scatter per pass).
// ---------------------------------------------------------------------------
__global__ void fill_buffers(float* __restrict__ smax,
                             float* __restrict__ den,
                             float* __restrict__ num, int n) {
    int i = blockIdx.x * blockDim.x + threadIdx.x;
    if (i < n) { smax[i] = -3.0e38f; den[i] = 0.f; num[i] = 0.f; }
}

__global__ void edge_max_kernel(const float* __restrict__ z,
                                const int* __restrict__ src,
                                const int* __restrict__ dst,
                                const float* __restrict__ tptr, int layer,
                                float* __restrict__ smax, int total) {
    int i = blockIdx.x * blockDim.x + threadIdx.x;
    if (i >= total) return;
    int e  = i >> 6;
    int ch = i & 63;
    float t   = tptr[layer];
    float msg = fmaxf(z[src[e] * Hc + ch], 0.f) + GEPS;
    atomicMaxF(&smax[dst[e] * Hc + ch], msg * t);
}

__global__ void edge_sum_kernel(const float* __restrict__ z,
                                const int* __restrict__ src,
                                const int* __restrict__ dst,
                                const float* __restrict__ tptr, int layer,
                                const float* __restrict__ smax,
                                float* __restrict__ den,
                                float* __restrict__ num, int total) {
    int i = blockIdx.x * blockDim.x + threadIdx.x;
    if (i >= total) return;
    int e  = i >> 6;
    int ch = i & 63;
    float t   = tptr[layer];
    float msg = fmaxf(z[src[e] * Hc + ch], 0.f) + GEPS;
    int   d   = dst[e] * Hc + ch;
    float ex  = __expf(msg * t - smax[d]);
    atomicAdd(&den[d], ex);
    atomicAdd(&num[d], msg * ex);
}

__global__ void agg_kernel(const float* __restrict__ num,
                           const float* __restrict__ den,
                           const float* __restrict__ z,
                           float* __restrict__ out, int n) {
    int i = blockIdx.x * blockDim.x + threadIdx.x;
    if (i < n) out[i] = num[i] / (den[i] + 1e-16f) + z[i];
}

__global__ void residual_kernel(float* __restrict__ h,
                                const float* __restrict__ y,
                                float beta, int n) {
    int i = blockIdx.x * blockDim.x + threadIdx.x;
    if (i < n) h[i] = beta * h[i] + y[i];
}

// ---------------------------------------------------------------------------
// log_softmax over rows of C=40 (thread-per-row; tiny output, 4.8 MB total).
// ---------------------------------------------------------------------------
__global__ void log_softmax_kernel(const float* __restrict__ X,
                                   float* __restrict__ Y, int rows, int C) {
    int r = blockIdx.x * blockDim.x + threadIdx.x;
    if (r >= rows) return;
    const float* x = X + (long)r * C;
    float m = -3.0e38f;
    for (int j = 0; j < C; ++j) m = fmaxf(m, x[j]);
    float s = 0.f;
    for (int j = 0; j < C; ++j) s += __expf(x[j] - m);
    float lse = m + __logf(s);
    float* y = Y + (long)r * C;
    for (int j = 0; j < C; ++j) y[j] = x[j] - lse;
}

// ---------------------------------------------------------------------------
// Host-side orchestration.
// ---------------------------------------------------------------------------
extern "C" void kernel_launch(void* const* d_in, const int* in_sizes, int n_in,
                              void* d_out, int out_size, void* d_ws, size_t ws_size,
                              hipStream_t stream) {
    (void)n_in; (void)out_size; (void)ws_size;

    const float* x    = (const float*)d_in[0];
    const int*   ei   = (const int*)  d_in[1];
    const float* encW = (const float*)d_in[2];
    const float* encB = (const float*)d_in[3];
    const float* t    = (const float*)d_in[4];
    const float* W1   = (const float*)d_in[5];
    const float* b1   = (const float*)d_in[6];
    const float* g1   = (const float*)d_in[7];
    const float* bb1  = (const float*)d_in[8];
    const float* W2   = (const float*)d_in[9];
    const float* b2   = (const float*)d_in[10];
    const float* ng   = (const float*)d_in[11];
    const float* nb   = (const float*)d_in[12];
    const float* linW = (const float*)d_in[13];
    const float* linB = (const float*)d_in[14];

    const int F = 128, H = 64, C = 40, L = 4;
    const int N = in_sizes[0] / F;
    const int E = in_sizes[1] / 2;
    const int* src = ei;
    const int* dst = ei + E;

    float* ws = (float*)d_ws;
    const long NH = (long)N * H;
    float* h    = ws;  ws += NH;
    float* z    = ws;  ws += NH;
    float* smax = ws;  ws += NH;
    float* den  = ws;  ws += NH;
    float* num  = ws;  ws += NH;
    float* outb = ws;  ws += NH;
    float* yb   = ws;  ws += NH;
    float* tmp  = ws;  ws += (long)N * 2 * H;   // reused for logits (N*C <= N*2H)

    const int  TB = 256;
    auto cdiv = [](long a, long b) { return (int)((a + b - 1) / b); };
    auto gemm_waves = [&](int M, int Nc) { return (long)(M / 16) * ((Nc + 63) / 64); };

    // encoder: h = x @ encW + encB   (K=128, N=64)
    gemm_bias_wmma<<<cdiv(gemm_waves(N, H) * 32, TB), TB, 0, stream>>>(x, encW, encB, h, N, H, F);

    const long totalE = (long)E * H;

    for (int l = 0; l < L; ++l) {
        const float* zin;
        if (l == 0) {
            zin = h;                                  // layer 0: conv applied to h directly
        } else {
            ln_relu_kernel<<<cdiv(N, 8), 256, 0, stream>>>(h, ng + l * H, nb + l * H, z, N, H);
            zin = z;
        }

        fill_buffers<<<cdiv(NH, TB), TB, 0, stream>>>(smax, den, num, (int)NH);
        edge_max_kernel<<<cdiv(totalE, TB), TB, 0, stream>>>(zin, src, dst, t, l, smax, (int)totalE);
        edge_sum_kernel<<<cdiv(totalE, TB), TB, 0, stream>>>(zin, src, dst, t, l, smax, den, num, (int)totalE);
        agg_kernel<<<cdiv(NH, TB), TB, 0, stream>>>(num, den, zin, outb, (int)NH);

        // MLP: tmp = outb @ W1[l] + b1[l]  -> LN+ReLU -> yb = tmp @ W2[l] + b2[l]
        gemm_bias_wmma<<<cdiv(gemm_waves(N, 2 * H) * 32, TB), TB, 0, stream>>>(
            outb, W1 + (long)l * H * 2 * H, b1 + l * 2 * H, tmp, N, 2 * H, H);
        ln_relu_kernel<<<cdiv(N, 8), 256, 0, stream>>>(tmp, g1 + l * 2 * H, bb1 + l * 2 * H, tmp, N, 2 * H);
        gemm_bias_wmma<<<cdiv(gemm_waves(N, H) * 32, TB), TB, 0, stream>>>(
            tmp, W2 + (long)l * 2 * H * H, b2 + l * H, yb, N, H, 2 * H);

        residual_kernel<<<cdiv(NH, TB), TB, 0, stream>>>(h, yb, (l == 0) ? 0.f : 1.f, (int)NH);
    }

    // final: z = relu(LN(h, norm_g[0], norm_b[0])); logits = z @ linW + linB; log_softmax
    ln_relu_kernel<<<cdiv(N, 8), 256, 0, stream>>>(h, ng, nb, z, N, H);
    gemm_bias_wmma<<<cdiv(gemm_waves(N, C) * 32, TB), TB, 0, stream>>>(z, linW, linB, tmp, N, C, H);
    log_softmax_kernel<<<cdiv(N, TB), TB, 0, stream>>>(tmp, (float*)d_out, N, C);
}